// CPFGNN_85246510891250
// MI455X (gfx1250) — compile-verified
//
#include <hip/hip_runtime.h>
#include <hip/hip_bf16.h>
#include <math.h>

typedef __attribute__((ext_vector_type(2))) float        v2f;
typedef __attribute__((ext_vector_type(8))) float        v8f;
typedef __attribute__((ext_vector_type(4))) unsigned int v4u;
typedef __attribute__((ext_vector_type(4))) int          v4i;
typedef __attribute__((ext_vector_type(8))) int          v8i;

#define F_IN 500
#define HID  64
#define C    10
#define RANK 3
#define KHOP 10
#define NK   (KHOP + 1)   // 11 Chebyshev terms
#define NKP  16           // padded to WMMA tile width

#if defined(__has_builtin)
#if __has_builtin(__builtin_amdgcn_wmma_f32_16x16x4_f32)
#define HAVE_WMMA_F32 1
#endif
#if __has_builtin(__builtin_amdgcn_tensor_load_to_lds) && \
    __has_builtin(__builtin_amdgcn_s_wait_tensorcnt)
#define HAVE_TDM 1
#endif
#endif

__device__ __forceinline__ int imin(int a, int b) { return a < b ? a : b; }

// ---------------- generic zero ----------------
__global__ void k_zero(float* __restrict__ p, int n) {
    int i = blockIdx.x * blockDim.x + threadIdx.x;
    if (i < n) p[i] = 0.f;
}

// ---------------- graph normalization ----------------
__global__ void k_deg(const int* __restrict__ edges, float* __restrict__ deg, int E) {
    int e = blockIdx.x * blockDim.x + threadIdx.x;
    if (e >= E) return;
    int s = edges[e], d = edges[E + e];
    if (s != d) atomicAdd(&deg[s], 1.0f);
}

__global__ void k_dinv(const float* __restrict__ deg, float* __restrict__ dinv, int n) {
    int i = blockIdx.x * blockDim.x + threadIdx.x;
    if (i < n) dinv[i] = (deg[i] > 0.f) ? rsqrtf(deg[i]) : 0.f;
}

// lam == 2 and diag == 0 always (w_off <= 0), so scaled w_off == raw w_off
__global__ void k_woff(const int* __restrict__ edges, const float* __restrict__ dinv,
                       float* __restrict__ woff, int E) {
    int e = blockIdx.x * blockDim.x + threadIdx.x;
    if (e >= E) return;
    int s = edges[e], d = edges[E + e];
    woff[e] = (s != d) ? (-dinv[s] * dinv[d]) : 0.f;
}

// ---------------- MLP ----------------
__global__ __launch_bounds__(HID) void k_mlp1(const float* __restrict__ feat,
                                              const float* __restrict__ W1,
                                              const float* __restrict__ b1,
                                              float* __restrict__ h, int n) {
    __shared__ float row[F_IN];
    int nid = blockIdx.x;
    int j = threadIdx.x;
    for (int f = j; f < F_IN; f += HID) row[f] = feat[(size_t)nid * F_IN + f];
    __syncthreads();
    float acc = b1[j];
#pragma clang loop unroll_count(4)
    for (int f = 0; f < F_IN; ++f) acc = fmaf(row[f], W1[f * HID + j], acc);
    h[(size_t)nid * HID + j] = fmaxf(acc, 0.f);
}

__global__ void k_mlp2(const float* __restrict__ h, const float* __restrict__ W2,
                       const float* __restrict__ b2, float* __restrict__ x, int n) {
    int i = blockIdx.x * blockDim.x + threadIdx.x;
    if (i >= n * C) return;
    int nid = i / C, c = i % C;
    float acc = b2[c];
    const float* hr = h + (size_t)nid * HID;
#pragma unroll
    for (int j = 0; j < HID; ++j) acc = fmaf(hr[j], W2[j * C + c], acc);
    x[i] = acc;
}

// ---------------- Chebyshev propagation ----------------
__global__ void k_scatter(const int* __restrict__ edges, const float* __restrict__ woff,
                          const float* __restrict__ t, float* __restrict__ P, int E) {
    int e = blockIdx.x * blockDim.x + threadIdx.x;
    if (e >= E) return;
    float w = woff[e];
    if (w == 0.f) return;
    int s = edges[e], d = edges[E + e];
    const float* ts = t + (size_t)s * C;
    float* pd = P + (size_t)d * C;
#pragma unroll
    for (int c = 0; c < C; ++c) atomicAdd(&pd[c], w * ts[c]);
}

__global__ void k_combine(const float* __restrict__ P, const float* __restrict__ Told,
                          float* __restrict__ Tnew, float alpha, float beta, int n10) {
    int i = blockIdx.x * blockDim.x + threadIdx.x;
    if (i < n10) Tnew[i] = fmaf(alpha, P[i], beta * Told[i]);
}

// ---------------- eta pre-mix: Emat[n][k] ----------------
__global__ void k_emat(const float* __restrict__ Tx, const float* __restrict__ Wp,
                       const float* __restrict__ bp, const float* __restrict__ gamma,
                       float* __restrict__ Emat, int n) {
    int i = blockIdx.x * blockDim.x + threadIdx.x;
    if (i >= n * NKP) return;
    int nid = i >> 4, k = i & 15;
    float e = 0.f;
    if (k < NK) {
        const float* t  = Tx + (size_t)k * n * C + (size_t)nid * C;
        const float* wp = Wp + k * C * RANK;
        float acc = 0.f;
#pragma unroll
        for (int r = 0; r < RANK; ++r) {
            float hr = bp[k * RANK + r];
#pragma unroll
            for (int c = 0; c < C; ++c) hr = fmaf(t[c], wp[c * RANK + r], hr);
            acc = fmaf(tanhf(hr), gamma[r * NK + k], acc);
        }
        e = acc * (1.f / (float)RANK);
    }
    Emat[i] = e;
}

// ---------------- Eta = CTC @ Emat  (n x n x 16), fp32 WMMA + split-K ----------------
#define WPB    4       // waves per block
#define KCHUNK 512     // B-chunk rows staged in LDS (512*16*4B = 32 KB)
#define SPLITK 8       // K-dimension slices (occupancy: 625*8 = 5000 waves)

#ifdef HAVE_TDM
// Issue one TDM DMA: B rows [kk0, kk0+chunk) x 16 f32 cols -> LDS at lds_byte_off.
// All operands are wave-uniform; EXEC is ignored by TENSOR_LOAD_TO_LDS.
__device__ __forceinline__ void tdm_stage(const float* B, int kk0, int chunk,
                                          unsigned lds_byte_off) {
    unsigned long long ga = (unsigned long long)(uintptr_t)(B + (size_t)kk0 * 16);
    v4u g0;
    g0[0] = 1u;                                             // count=1 (valid user D#)
    g0[1] = lds_byte_off;                                   // lds_addr
    g0[2] = (unsigned)(ga & 0xFFFFFFFFu);                   // global_addr[31:0]
    g0[3] = (unsigned)((ga >> 32) & 0x01FFFFFFu) | (2u << 30); // addr[56:32] | type=2
    v8i g1;
    g1[0] = (2 << 16);                                      // data_size = 4B dwords
    g1[1] = (16 << 16);                                     // tensor_dim0 = 16 (lo bits)
    g1[2] = (chunk & 0xFFFF) << 16;                         // dim0 hi=0 | tensor_dim1 lo
    g1[3] = ((chunk >> 16) & 0xFFFF) | (16 << 16);          // tensor_dim1 hi | tile_dim0=16
    g1[4] = (chunk & 0xFFFF);                               // tile_dim1 = chunk rows
    g1[5] = 16;                                             // tensor_dim0_stride = 16
    g1[6] = 0;
    g1[7] = 0;
    v4i z4 = {0, 0, 0, 0};
    v8i z8 = {0, 0, 0, 0, 0, 0, 0, 0};
    // 6-arg form (clang-23 / therock headers): (g0, g1, g2, g3, g4, cpol)
    __builtin_amdgcn_tensor_load_to_lds(g0, g1, z4, z4, z8, 0);
}
#endif

#ifdef HAVE_WMMA_F32
__global__ __launch_bounds__(WPB * 32) void k_gemm_wmma(const float* __restrict__ A,
                                                        const float* __restrict__ B,
                                                        float* __restrict__ D, int n) {
#ifdef HAVE_TDM
    __shared__ __align__(16) float Bs[2 * KCHUNK * 16];     // 64 KB, double buffered
#else
    __shared__ __align__(16) float Bs[KCHUNK * 16];
#endif
    const int wave = threadIdx.x >> 5;
    const int lane = threadIdx.x & 31;
    const int half = lane >> 4;
    const int col  = lane & 15;
    const int m0   = (blockIdx.x * WPB + wave) * 16;
    // clamp: padded lanes read row n-1 and produce garbage only in unsaved rows
    const int row  = imin(m0 + col, n - 1);
    // 4-aligned K-slice for this block (split-K)
    const int per  = (((n + SPLITK - 1) / SPLITK) + 3) & ~3;
    const int kbeg = blockIdx.y * per;
    const int kend = imin(n, kbeg + per);
    const int nch  = (kend > kbeg) ? (kend - kbeg + KCHUNK - 1) / KCHUNK : 0;
    const float* Arow = A + (size_t)row * n + 2 * half;

    v8f acc = {};
#ifdef HAVE_TDM
    const bool w0 = (threadIdx.x < 32);
    const unsigned lds0 = (unsigned)(uintptr_t)&Bs[0];
    const unsigned lds1 = (unsigned)(uintptr_t)&Bs[KCHUNK * 16];
    if (nch > 0 && w0) tdm_stage(B, kbeg, imin(KCHUNK, kend - kbeg), lds0);
    int cur = 0;
    for (int i = 0; i < nch; ++i) {
        const int kk0   = kbeg + i * KCHUNK;
        const int chunk = imin(KCHUNK, kend - kk0);
        if (w0) {
            if (i + 1 < nch) {
                const int kk1 = kk0 + KCHUNK;
                tdm_stage(B, kk1, imin(KCHUNK, kend - kk1), cur ? lds0 : lds1);
                __builtin_amdgcn_s_wait_tensorcnt((short)1);  // chunk i landed
            } else {
                __builtin_amdgcn_s_wait_tensorcnt((short)0);
            }
        }
        __syncthreads();
        const float* Bsbuf = Bs + (cur ? KCHUNK * 16 : 0);
#pragma clang loop unroll_count(4)
        for (int k = 0; k < chunk; k += 4) {
            float2 t = *(const float2*)(Arow + kk0 + k);     // lanes: K pairs per layout
            v2f a; a.x = t.x; a.y = t.y;
            int base = (k + 2 * half) * 16 + col;
            v2f b; b.x = Bsbuf[base]; b.y = Bsbuf[base + 16];
            acc = __builtin_amdgcn_wmma_f32_16x16x4_f32(false, a, false, b,
                                                        (short)0, acc, false, false);
        }
        __syncthreads();
        cur ^= 1;
    }
#else
    for (int i = 0; i < nch; ++i) {
        const int kk0   = kbeg + i * KCHUNK;
        const int chunk = imin(KCHUNK, kend - kk0);
        for (int idx = threadIdx.x * 4; idx < KCHUNK * 16; idx += blockDim.x * 4) {
            int r = idx >> 4;
            float4 v = make_float4(0.f, 0.f, 0.f, 0.f);
            if (r < chunk) v = *(const float4*)(B + (size_t)(kk0 + r) * 16 + (idx & 15));
            *(float4*)(Bs + idx) = v;
        }
        __syncthreads();
#pragma clang loop unroll_count(4)
        for (int k = 0; k < chunk; k += 4) {
            float2 t = *(const float2*)(Arow + kk0 + k);
            v2f a; a.x = t.x; a.y = t.y;
            int base = (k + 2 * half) * 16 + col;
            v2f b; b.x = Bs[base]; b.y = Bs[base + 16];
            acc = __builtin_amdgcn_wmma_f32_16x16x4_f32(false, a, false, b,
                                                        (short)0, acc, false, false);
        }
        __syncthreads();
    }
#endif
    // C/D layout: VGPR r -> M = m0 + r + 8*half, N = col; combine split-K partials
#pragma unroll
    for (int r = 0; r < 8; ++r) {
        int orow = m0 + r + 8 * half;
        if (orow < n) atomicAdd(&D[(size_t)orow * 16 + col], acc[r]);
    }
}
#else
__global__ void k_gemm_fallback(const float* __restrict__ A, const float* __restrict__ B,
                                float* __restrict__ D, int n) {
    int i = blockIdx.x * blockDim.x + threadIdx.x;
    if (i >= n * 16) return;
    int nid = i >> 4, k = i & 15;
    float acc = 0.f;
    const float* ar = A + (size_t)nid * n;
    for (int j = 0; j < n; ++j) acc = fmaf(ar[j], B[(size_t)j * 16 + k], acc);
    D[i] = acc;
}
#endif

// ---------------- hidden accumulate + log_softmax ----------------
__global__ void k_final(const float* __restrict__ Tx, const float* __restrict__ Eta,
                        float* __restrict__ out, int n) {
    int nid = blockIdx.x * blockDim.x + threadIdx.x;
    if (nid >= n) return;
    float hsum[C];
#pragma unroll
    for (int c = 0; c < C; ++c) hsum[c] = 0.f;
#pragma unroll
    for (int k = 0; k < NK; ++k) {
        float ek = Eta[(size_t)nid * 16 + k];
        const float* t = Tx + (size_t)k * n * C + (size_t)nid * C;
#pragma unroll
        for (int c = 0; c < C; ++c) hsum[c] = fmaf(t[c], ek, hsum[c]);
    }
    float m = hsum[0];
#pragma unroll
    for (int c = 1; c < C; ++c) m = fmaxf(m, hsum[c]);
    float s = 0.f;
#pragma unroll
    for (int c = 0; c < C; ++c) s += expf(hsum[c] - m);
    float ls = logf(s);
#pragma unroll
    for (int c = 0; c < C; ++c) out[(size_t)nid * C + c] = hsum[c] - m - ls;
}

extern "C" void kernel_launch(void* const* d_in, const int* in_sizes, int n_in,
                              void* d_out, int out_size, void* d_ws, size_t ws_size,
                              hipStream_t stream) {
    const float* feat  = (const float*)d_in[0];
    const int*   edges = (const int*)d_in[1];   // JAX x64 disabled -> int32
    const float* CTC   = (const float*)d_in[2];
    const float* W1    = (const float*)d_in[3];
    const float* b1    = (const float*)d_in[4];
    const float* W2    = (const float*)d_in[5];
    const float* b2    = (const float*)d_in[6];
    const float* gam   = (const float*)d_in[7];
    const float* Wp    = (const float*)d_in[8];
    const float* bp    = (const float*)d_in[9];
    float* out = (float*)d_out;

    const int n = in_sizes[0] / F_IN;
    const int E = in_sizes[1] / 2;

    float* ws   = (float*)d_ws;
    float* deg  = ws;  ws += n;
    float* dinv = ws;  ws += n;
    float* woff = ws;  ws += E;
    float* h    = ws;  ws += (size_t)n * HID;
    float* Tx   = ws;  ws += (size_t)NK * n * C;
    float* P    = ws;  ws += (size_t)n * C;
    float* Emat = ws;  ws += (size_t)n * NKP;
    float* Eta  = ws;  ws += (size_t)n * NKP;

    const int B = 256;
    k_zero<<<(n + B - 1) / B, B, 0, stream>>>(deg, n);
    k_deg<<<(E + B - 1) / B, B, 0, stream>>>(edges, deg, E);
    k_dinv<<<(n + B - 1) / B, B, 0, stream>>>(deg, dinv, n);
    k_woff<<<(E + B - 1) / B, B, 0, stream>>>(edges, dinv, woff, E);

    k_mlp1<<<n, HID, 0, stream>>>(feat, W1, b1, h, n);
    k_mlp2<<<(n * C + B - 1) / B, B, 0, stream>>>(h, W2, b2, Tx, n);

    for (int k = 0; k < KHOP; ++k) {
        k_zero<<<(n * C + B - 1) / B, B, 0, stream>>>(P, n * C);
        k_scatter<<<(E + B - 1) / B, B, 0, stream>>>(edges, woff,
                                                     Tx + (size_t)k * n * C, P, E);
        const float* Told = (k == 0) ? P : Tx + (size_t)(k - 1) * n * C;
        float alpha = (k == 0) ? 1.f : 2.f;
        float beta  = (k == 0) ? 0.f : -1.f;
        k_combine<<<(n * C + B - 1) / B, B, 0, stream>>>(P, Told,
                                                         Tx + (size_t)(k + 1) * n * C,
                                                         alpha, beta, n * C);
    }

    k_emat<<<(n * NKP + B - 1) / B, B, 0, stream>>>(Tx, Wp, bp, gam, Emat, n);
#ifdef HAVE_WMMA_F32
    k_zero<<<(n * NKP + B - 1) / B, B, 0, stream>>>(Eta, n * NKP);
    int mt = (n + 15) / 16;
    dim3 grid((mt + WPB - 1) / WPB, SPLITK, 1);
    k_gemm_wmma<<<grid, WPB * 32, 0, stream>>>(CTC, Emat, Eta, n);
#else
    k_gemm_fallback<<<(n * 16 + B - 1) / B, B, 0, stream>>>(CTC, Emat, Eta, n);
#endif
    k_final<<<(n + B - 1) / B, B, 0, stream>>>(Tx, Eta, out, n);
}